// BehlerG2_62328565399850
// MI455X (gfx1250) — compile-verified
//
#include <hip/hip_runtime.h>

typedef __attribute__((ext_vector_type(2))) float v2f;
typedef __attribute__((ext_vector_type(8))) float v8f;

__device__ __forceinline__ float cosine_cutoff(float r) {
    // 0.5*(cos(pi*r/CUTOFF)+1) * (r < CUTOFF), CUTOFF = 6
    float c = 0.5f * (__cosf(r * (3.14159265358979323846f / 6.0f)) + 1.0f);
    return (r < 6.0f) ? c : 0.0f;
}

// One wave handles TWO atoms (an "atom pair"): WMMA f32 16x16x4 with
//   A rows 0-7  = atom0 eta rows, rows 8-15 = atom1 eta rows, K = 4 t-values
//   B cols 0-7  = atom0 zeta cols, cols 8-15 = atom1 zeta cols
// D diagonal 8x8 blocks = per-atom f[e,z]; off-diagonal blocks unused.
__global__ __launch_bounds__(256) void behler_angular_wmma(
    const float* __restrict__ positions,  // B,A,3
    const float* __restrict__ cell,       // B,3,3
    const int*   __restrict__ nj,         // B,A,T
    const int*   __restrict__ nk,         // B,A,T
    const int*   __restrict__ maskp,      // B,A,T
    const float* __restrict__ offj,       // B,A,T,3
    const float* __restrict__ offk,       // B,A,T,3
    const int*   __restrict__ zatom,      // B,A
    const float* __restrict__ etas,       // 8
    float* __restrict__ out,              // B,A,64  (e major, z minor)
    int B, int A, int T)
{
    const int lane = threadIdx.x & 31;
    const int wave = threadIdx.x >> 5;
    const int pairsPerB = A >> 1;
    const int pair = blockIdx.x * (blockDim.x >> 5) + wave;
    if (pair >= B * pairsPerB) return;           // uniform per wave (grid sized exactly)

    const int b  = pair / pairsPerB;
    const int a0 = (pair % pairsPerB) << 1;

    // f32 16x16x4 fragment ownership for this lane:
    const int idx8    = lane & 7;                // eta row index == zeta col index
    const int atomSel = (lane >> 3) & 1;         // (lane%16)/8 -> which atom of the pair
    const int a       = a0 + atomSel;
    const int kBase   = (lane >> 4) << 1;        // lane needs K = kBase, kBase+1

    const float eta = etas[idx8];
    const int   nsq = idx8 & 3;                  // zeta = 1<<nsq -> nsq squarings

    // hoist pos_i and cell[b]
    float Pi[3], C[9];
    #pragma unroll
    for (int d = 0; d < 3; ++d) Pi[d] = positions[((size_t)(b * A + a)) * 3 + d];
    #pragma unroll
    for (int q = 0; q < 9; ++q) C[q] = cell[(size_t)b * 9 + q];

    const size_t rowBase = ((size_t)(b * A + a)) * (size_t)T;   // triple row for atom a
    const size_t posBase = ((size_t)b) * A * 3;
    const size_t zBase   = ((size_t)b) * A;

    v8f acc = {};

    for (int tc = 0; tc < T; tc += 4) {
        v2f Av, Bv;
        #pragma unroll
        for (int s2 = 0; s2 < 2; ++s2) {
            const int t = tc + kBase + s2;
            const size_t ti = rowBase + (size_t)t;
            const int j = nj[ti];
            const int k = nk[ti];
            const int m = maskp[ti];

            // Unconditional loads: neighbor indices are always in [0, A), so no
            // EXEC-divergent branch is needed; mask is applied via selects only.
            const float zj = (float)zatom[zBase + j];
            const float zk = (float)zatom[zBase + k];

            const float oj0 = offj[ti * 3 + 0], oj1 = offj[ti * 3 + 1], oj2 = offj[ti * 3 + 2];
            const float ok0 = offk[ti * 3 + 0], ok1 = offk[ti * 3 + 1], ok2 = offk[ti * 3 + 2];

            float d2ij = 0.f, d2ik = 0.f, d2jk = 0.f;
            #pragma unroll
            for (int d = 0; d < 3; ++d) {
                const float pj = positions[posBase + (size_t)j * 3 + d]
                               + oj0 * C[0 * 3 + d] + oj1 * C[1 * 3 + d] + oj2 * C[2 * 3 + d];
                const float pk = positions[posBase + (size_t)k * 3 + d]
                               + ok0 * C[0 * 3 + d] + ok1 * C[1 * 3 + d] + ok2 * C[2 * 3 + d];
                const float eij = pj - Pi[d];
                const float eik = pk - Pi[d];
                const float ejk = pj - pk;
                d2ij += eij * eij;
                d2ik += eik * eik;
                d2jk += ejk * ejk;
            }

            const bool msk = (m != 0);
            const float r2ij = (msk ? d2ij : 1.0f) + 1e-12f;   // safe_norm semantics
            const float r2ik = (msk ? d2ik : 1.0f) + 1e-12f;
            const float r2jk = (msk ? d2jk : 1.0f) + 1e-12f;
            const float rij = __builtin_sqrtf(r2ij);
            const float rik = __builtin_sqrtf(r2ik);
            const float rjk = __builtin_sqrtf(r2jk);

            const float cut = cosine_cutoff(rij) * cosine_cutoff(rik) * cosine_cutoff(rjk);
            const float w   = msk ? (zj * zk) : 0.0f;

            // A-side value: radial * cutoff * (z_j*z_k*mask)   (exact f32)
            const float aval = __expf(-eta * (r2ij + r2ik + r2jk)) * cut * w;

            // B-side value: (1 - cos_theta)^zeta via repeated squaring.
            // 1/(2*rij*rik) == 0.5 * rsq(r2ij*r2ik): one v_rsq_f32 instead of a
            // full IEEE divide expansion.
            const float inv2r = 0.5f * __builtin_amdgcn_rsqf(r2ij * r2ik);
            const float ct = msk ? ((r2ij + r2ik - r2jk) * inv2r) : 0.0f;
            const float p1 = 1.0f - ct;
            const float p2 = p1 * p1;
            const float p4 = p2 * p2;
            const float p8 = p4 * p4;
            const float bval = (nsq == 0) ? p1 : (nsq == 1) ? p2 : (nsq == 2) ? p4 : p8;

            if (s2 == 0) { Av.x = aval; Bv.x = bval; }
            else         { Av.y = aval; Bv.y = bval; }
        }
        // D = A(16x4) * B(4x16) + D, exact f32 accumulation
        acc = __builtin_amdgcn_wmma_f32_16x16x4_f32(
            /*neg_a=*/false, Av, /*neg_b=*/false, Bv,
            /*c_mod=*/(short)0, acc, /*reuse_a=*/false, /*reuse_b=*/false);
    }

    // C/D layout: VGPR r -> (M=r, N=lane) for lanes 0-15, (M=r+8, N=lane-16) for lanes 16-31.
    // Diagonal blocks: lanes 0-7  -> atom0, e=r,   z=lane
    //                  lanes 24-31-> atom1, e=r,   z=lane-24
    int zi = -1, ao = a0;
    if (lane < 8)        { zi = lane;       ao = a0;     }
    else if (lane >= 24) { zi = lane - 24;  ao = a0 + 1; }
    if (zi >= 0) {
        const int zexp = 1 << (zi & 3);
        const float pf = exp2f((float)((zi < 4) ? (1 - zexp) : (1 + zexp)));  // 2^(1∓ζ)
        float* o = out + ((size_t)(b * A + ao)) * 64 + zi;
        #pragma unroll
        for (int r = 0; r < 8; ++r) o[r * 8] = acc[r] * pf;
    }
}

extern "C" void kernel_launch(void* const* d_in, const int* in_sizes, int n_in,
                              void* d_out, int out_size, void* d_ws, size_t ws_size,
                              hipStream_t stream) {
    const float* positions = (const float*)d_in[0];
    const float* cell      = (const float*)d_in[1];
    const int*   nj        = (const int*)d_in[2];
    const int*   nk        = (const int*)d_in[3];
    const int*   maskp     = (const int*)d_in[4];
    // d_in[5] = offsets: unused by the reference computation
    const float* offj      = (const float*)d_in[6];
    const float* offk      = (const float*)d_in[7];
    const int*   zatom     = (const int*)d_in[8];
    const float* etas      = (const float*)d_in[9];
    float* out = (float*)d_out;

    const int B = in_sizes[1] / 9;                 // cell is (B,3,3)
    const int A = in_sizes[8] / B;                 // atomic_numbers is (B,A)
    const int T = in_sizes[2] / (B * A);           // neighbors_j is (B,A,T)

    const int pairs = (B * A) / 2;                 // one wave per 2 atoms
    const int wavesPerBlock = 8;                   // 256 threads = 8 wave32
    const int blocks = (pairs + wavesPerBlock - 1) / wavesPerBlock;

    behler_angular_wmma<<<blocks, wavesPerBlock * 32, 0, stream>>>(
        positions, cell, nj, nk, maskp, offj, offk, zatom, etas, out, B, A, T);
}